// GINEncoder_33225867002147
// MI455X (gfx1250) — compile-verified
//
#include <hip/hip_runtime.h>
#include <cstddef>

// ---------------- problem constants ----------------
#define GIN_N     50000
#define GIN_E     800000
#define GIN_D     128
#define GIN_L     3
#define GIN_EPS   1e-5f

// ---------------- tiling constants -----------------
#define GIN_WAVES 8                  // waves per block (wave32)
#define GIN_ROWS  (GIN_WAVES * 16)   // rows per block = 128
#define GIN_WP    288                // floats per pair-row of swizzled W (==32 mod 64 -> conflict-free b64)
#define GIN_ASTR  132                // padded LDS row stride (floats) for A staging

typedef __attribute__((ext_vector_type(2))) float v2f;
typedef __attribute__((ext_vector_type(8))) float v8f;

// ============================================================
// pre[i] = h[i]   (vectorized copy, 16B)
// ============================================================
__global__ __launch_bounds__(256)
void gin_copy_kernel(const float4* __restrict__ in, float4* __restrict__ out, int n4) {
    int i = blockIdx.x * 256 + threadIdx.x;
    if (i < n4) out[i] = in[i];
}

// ============================================================
// pre[dst] += h[src]  — one wave per edge, one float4 per lane.
// Gather is L2-resident (h is 25.6MB << 192MB L2); atomics are
// non-returning global_atomic_add_f32.
// ============================================================
__global__ __launch_bounds__(256)
void gin_scatter_kernel(const float* __restrict__ h,
                        const int* __restrict__ src,
                        const int* __restrict__ dst,
                        float* __restrict__ agg, int ne) {
    int t = blockIdx.x * 256 + threadIdx.x;
    int e = t >> 5;
    if (e >= ne) return;
    int lane = t & 31;
    const float4 v = *(const float4*)(h + (size_t)src[e] * GIN_D + lane * 4);
    float* p = agg + (size_t)dst[e] * GIN_D + lane * 4;
    __hip_atomic_fetch_add(p + 0, v.x, __ATOMIC_RELAXED, __HIP_MEMORY_SCOPE_AGENT);
    __hip_atomic_fetch_add(p + 1, v.y, __ATOMIC_RELAXED, __HIP_MEMORY_SCOPE_AGENT);
    __hip_atomic_fetch_add(p + 2, v.z, __ATOMIC_RELAXED, __HIP_MEMORY_SCOPE_AGENT);
    __hip_atomic_fetch_add(p + 3, v.w, __ATOMIC_RELAXED, __HIP_MEMORY_SCOPE_AGENT);
}

// ============================================================
// Fused  out = ReLU(BN2( ReLU(BN1(pre @ W1 + b1)) @ W2 + b2 ))
// One 8-wave block per 128 rows; V_WMMA_F32_16X16X4_F32 tiles.
// Weights live in LDS pre-paired: wp[k/2][n] = {W[k][n], W[k+1][n]},
// so each B fragment is ONE aligned ds_load_b64 (no v_mov marshaling).
// ============================================================
__global__ __launch_bounds__(256)
void gin_mlp_kernel(const float* __restrict__ hin,
                    const float* __restrict__ W1, const float* __restrict__ b1,
                    const float* __restrict__ W2, const float* __restrict__ b2,
                    const float* __restrict__ g0, const float* __restrict__ be0,
                    const float* __restrict__ rm0, const float* __restrict__ rv0,
                    const float* __restrict__ g1, const float* __restrict__ be1,
                    const float* __restrict__ rm1, const float* __restrict__ rv1,
                    float* __restrict__ out, int nrows) {
    extern __shared__ float smem[];
    float* w1s = smem;                              // 64 pair-rows * GIN_WP
    float* w2s = w1s + 64 * GIN_WP;
    float* stg = w2s + 64 * GIN_WP;                 // GIN_WAVES*16*GIN_ASTR
    float* S1  = stg + GIN_WAVES * 16 * GIN_ASTR;   // 128
    float* T1  = S1 + 128;
    float* S2  = T1 + 128;
    float* T2  = S2 + 128;

    const int tid = threadIdx.x;

    // ---- cooperative preamble: pair-swizzled weights + folded BN params ----
    for (int i = tid; i < 128 * 128; i += 256) {
        int k = i >> 7, n = i & 127;
        int off = (k >> 1) * GIN_WP + 2 * n + (k & 1);
        w1s[off] = W1[i];
        w2s[off] = W2[i];
    }
    if (tid < 128) {
        float s1 = g0[tid] * rsqrtf(rv0[tid] + GIN_EPS);
        S1[tid] = s1;
        T1[tid] = (b1[tid] - rm0[tid]) * s1 + be0[tid];
        float s2 = g1[tid] * rsqrtf(rv1[tid] + GIN_EPS);
        S2[tid] = s2;
        T2[tid] = (b2[tid] - rm1[tid]) * s2 + be1[tid];
    }
    __syncthreads();

    const int wave = tid >> 5;
    const int lane = tid & 31;
    const int lrow = lane & 15;   // M (for A) / N (for B) within tile
    const int half = lane >> 4;   // K-half selector per ISA f32 layouts
    float* myst = stg + wave * 16 * GIN_ASTR;

    const int r0 = blockIdx.x * GIN_ROWS + wave * 16;

    // ---- stage this wave's 16 input rows into padded LDS stripe ----
    #pragma unroll 4
    for (int r = 0; r < 16; ++r) {
        int gr = r0 + r;
        if (gr >= nrows) gr = nrows - 1;  // clamp: keep EXEC uniform
        const float4 v = *(const float4*)(hin + (size_t)gr * GIN_D + lane * 4);
        float* d = &myst[r * GIN_ASTR + lane * 4];
        d[0] = v.x; d[1] = v.y; d[2] = v.z; d[3] = v.w;
    }

    // ---- GEMM 1: acc[nt] (16x16 f32 tiles) = A(16x128) @ W1 ----
    v8f acc[8] = {};
    for (int k0 = 0; k0 < 128; k0 += 4) {
        const int ka = k0 + 2 * half;
        // A fragment (ISA 16x4 f32 layout): lane holds A[lrow][ka], A[lrow][ka+1]
        const float2 af = *(const float2*)&myst[lrow * GIN_ASTR + ka];
        v2f a; a.x = af.x; a.y = af.y;
        const float* wrow = &w1s[(size_t)((k0 >> 1) + half) * GIN_WP + 2 * lrow];
        #pragma unroll
        for (int nt = 0; nt < 8; ++nt) {
            const float2 bf = *(const float2*)&wrow[nt * 32];  // {W[ka][n], W[ka+1][n]}
            v2f b; b.x = bf.x; b.y = bf.y;
            acc[nt] = __builtin_amdgcn_wmma_f32_16x16x4_f32(
                false, a, false, b, (short)0, acc[nt], false, false);
        }
    }

    // ---- epilogue 1: fused bias+BN+ReLU, restage z into same stripe ----
    // (same-wave LDS is in-order: WAR vs A reads and RAW vs GEMM2 reads are safe)
    #pragma unroll
    for (int nt = 0; nt < 8; ++nt) {
        const int n = nt * 16 + lrow;
        const float s = S1[n], t = T1[n];
        #pragma unroll
        for (int v = 0; v < 8; ++v) {
            const int m = v + 8 * half;   // C/D layout: lane>=16 holds M=v+8
            myst[m * GIN_ASTR + n] = fmaxf(acc[nt][v] * s + t, 0.0f);
        }
    }

    // ---- GEMM 2: acc2 = Z(16x128) @ W2 ----
    v8f acc2[8] = {};
    for (int k0 = 0; k0 < 128; k0 += 4) {
        const int ka = k0 + 2 * half;
        const float2 af = *(const float2*)&myst[lrow * GIN_ASTR + ka];
        v2f a; a.x = af.x; a.y = af.y;
        const float* wrow = &w2s[(size_t)((k0 >> 1) + half) * GIN_WP + 2 * lrow];
        #pragma unroll
        for (int nt = 0; nt < 8; ++nt) {
            const float2 bf = *(const float2*)&wrow[nt * 32];
            v2f b; b.x = bf.x; b.y = bf.y;
            acc2[nt] = __builtin_amdgcn_wmma_f32_16x16x4_f32(
                false, a, false, b, (short)0, acc2[nt], false, false);
        }
    }

    // ---- epilogue 2: fused bias+BN+ReLU, predicated global store ----
    #pragma unroll
    for (int nt = 0; nt < 8; ++nt) {
        const int n = nt * 16 + lrow;
        const float s = S2[n], t = T2[n];
        #pragma unroll
        for (int v = 0; v < 8; ++v) {
            const int gr = r0 + v + 8 * half;
            if (gr < nrows)
                out[(size_t)gr * GIN_D + n] = fmaxf(acc2[nt][v] * s + t, 0.0f);
        }
    }
}

// ============================================================
extern "C" void kernel_launch(void* const* d_in, const int* in_sizes, int n_in,
                              void* d_out, int out_size, void* d_ws, size_t ws_size,
                              hipStream_t stream) {
    const float* x  = (const float*)d_in[0];
    const int*   ei = (const int*)d_in[1];
    const float* W1 = (const float*)d_in[2];
    const float* b1 = (const float*)d_in[3];
    const float* W2 = (const float*)d_in[4];
    const float* b2 = (const float*)d_in[5];
    const float* g  = (const float*)d_in[6];
    const float* be = (const float*)d_in[7];
    const float* rm = (const float*)d_in[8];
    const float* rv = (const float*)d_in[9];
    float* out = (float*)d_out;

    const int   nn  = GIN_N;
    const int   ne  = GIN_E;
    const int*  src = ei;
    const int*  dst = ei + ne;

    float* pre  = (float*)d_ws;                 // N*D  aggregation buffer
    float* hbuf = pre + (size_t)nn * GIN_D;     // N*D  inter-layer features

    const size_t smem_bytes =
        (size_t)(2 * 64 * GIN_WP + GIN_WAVES * 16 * GIN_ASTR + 4 * 128) * sizeof(float);
    (void)hipFuncSetAttribute((const void*)gin_mlp_kernel,
                              hipFuncAttributeMaxDynamicSharedMemorySize,
                              (int)smem_bytes);

    const int copy_n4     = nn * GIN_D / 4;
    const int copy_blocks = (copy_n4 + 255) / 256;
    const int scat_blocks = (ne * 32 + 255) / 256;
    const int mlp_blocks  = (nn + GIN_ROWS - 1) / GIN_ROWS;

    for (int i = 0; i < GIN_L; ++i) {
        const float* hin  = (i == 0) ? x : hbuf;
        float*       hout = (i == GIN_L - 1) ? out : hbuf;

        gin_copy_kernel<<<copy_blocks, 256, 0, stream>>>(
            (const float4*)hin, (float4*)pre, copy_n4);
        gin_scatter_kernel<<<scat_blocks, 256, 0, stream>>>(
            hin, src, dst, pre, ne);
        gin_mlp_kernel<<<mlp_blocks, 256, smem_bytes, stream>>>(
            pre,
            W1 + (size_t)i * GIN_D * GIN_D, b1 + (size_t)i * GIN_D,
            W2 + (size_t)i * GIN_D * GIN_D, b2 + (size_t)i * GIN_D,
            g  + (size_t)(i * 2 + 0) * GIN_D, be + (size_t)(i * 2 + 0) * GIN_D,
            rm + (size_t)(i * 2 + 0) * GIN_D, rv + (size_t)(i * 2 + 0) * GIN_D,
            g  + (size_t)(i * 2 + 1) * GIN_D, be + (size_t)(i * 2 + 1) * GIN_D,
            rm + (size_t)(i * 2 + 1) * GIN_D, rv + (size_t)(i * 2 + 1) * GIN_D,
            hout, nn);
    }
}